// MessagePassing_11433202942341
// MI455X (gfx1250) — compile-verified
//
#include <hip/hip_runtime.h>

// MessagePassing(aggr='add'), identity message:
//   h[row[e], :] += x[col[e], :]  for e in [0, E)
// x: [N=10000, F=128] fp32 (5.12 MB -> fully L2-resident in MI455X's 192MB L2)
// edge_index: [2, E=640000] int32 (rows = ptr[0:E], cols = ptr[E:2E])
// out: [N, F] fp32
//
// Strategy (avg degree E/N = 64 => atomics are 64x redundant work):
//   Build CSR-by-destination with a counting sort (int atomics + one-block scan),
//   then a destination-centric gather-sum: one wave per node, register
//   accumulation, ZERO fp atomics, one b128 store per lane. The per-node column
//   list is streamed through a double-buffered global_load_async_to_lds_b32
//   pipeline (gfx1250 ASYNCcnt path) so index fetch overlaps the x-row gathers.
//
// Fallback (if d_ws too small): round-1 atomic scatter with async-LDS staging
// and hardware global_atomic_add_f32.

#define N_FEAT 128
#define WPB    8     // waves per 256-thread block (wave32)
#define TILE   256   // edges per block in the fallback scatter

// ---------------------------------------------------------------------------
// CSR-build kernels
// ---------------------------------------------------------------------------

__global__ __launch_bounds__(256) void zero_cnt_kernel(int* __restrict__ cnt, int n) {
    for (int i = blockIdx.x * blockDim.x + threadIdx.x; i < n;
         i += gridDim.x * blockDim.x)
        cnt[i] = 0;
}

__global__ __launch_bounds__(256) void count_kernel(const int* __restrict__ rows,
                                                    int* __restrict__ cnt, int n_edges) {
    for (int i = blockIdx.x * blockDim.x + threadIdx.x; i < n_edges;
         i += gridDim.x * blockDim.x)
        atomicAdd(&cnt[rows[i]], 1);           // global_atomic_add_u32 (no return)
}

// Single-block exclusive scan of cnt[0..n) -> offs[0..n], offs[n] = total.
// Also rewrites cnt[i] with the exclusive prefix (cursor for the bucket pass).
__global__ __launch_bounds__(1024) void scan_kernel(int* __restrict__ cnt,
                                                    int* __restrict__ offs,
                                                    int n, int total) {
    __shared__ int sdata[1024];
    __shared__ int s_carry;
    const int tid = threadIdx.x;
    if (tid == 0) s_carry = 0;
    __syncthreads();

    for (int base = 0; base < n; base += 1024) {
        int i = base + tid;
        int v = (i < n) ? cnt[i] : 0;
        sdata[tid] = v;
        __syncthreads();
        // Hillis-Steele inclusive scan over 1024 elements in LDS
        for (int d = 1; d < 1024; d <<= 1) {
            int t = (tid >= d) ? sdata[tid - d] : 0;
            __syncthreads();
            sdata[tid] += t;
            __syncthreads();
        }
        int incl  = sdata[tid];
        int excl  = incl - v;
        int carry = s_carry;                   // read before 1023 updates it
        if (i < n) {
            int e = carry + excl;
            offs[i] = e;
            cnt[i]  = e;                       // cursor copy (aliases cnt)
        }
        __syncthreads();
        if (tid == 1023) s_carry = carry + incl;
        __syncthreads();
    }
    if (tid == 0) offs[n] = total;
}

__global__ __launch_bounds__(256) void bucket_kernel(const int* __restrict__ rows,
                                                     const int* __restrict__ cols,
                                                     int* __restrict__ cursor,
                                                     int* __restrict__ bucket,
                                                     int n_edges) {
    for (int i = blockIdx.x * blockDim.x + threadIdx.x; i < n_edges;
         i += gridDim.x * blockDim.x) {
        int p = atomicAdd(&cursor[rows[i]], 1); // global_atomic_add_u32 (return)
        bucket[p] = cols[i];
    }
}

// ---------------------------------------------------------------------------
// Destination-centric gather-sum: one wave per node, no fp atomics.
// Column list streamed via double-buffered async-to-LDS copies.
// ---------------------------------------------------------------------------

__global__ __launch_bounds__(256) void gather_sum_kernel(
        const float* __restrict__ x,
        const int*   __restrict__ offs,    // [N+1]
        const int*   __restrict__ bucket,  // [E] col indices sorted by dest
        float*       __restrict__ out,
        int n_nodes) {
    __shared__ int s_cols[WPB][2][32];     // per-wave double buffer
    const int lane = threadIdx.x & 31;
    const int wave = threadIdx.x >> 5;

    const unsigned lds0 = (unsigned)(size_t)&s_cols[wave][0][lane];
    const unsigned lds1 = (unsigned)(size_t)&s_cols[wave][1][lane];

    for (int node = blockIdx.x * WPB + wave; node < n_nodes;
         node += gridDim.x * WPB) {
        const int beg = offs[node];
        const int end = offs[node + 1];
        const int deg = end - beg;

        float4 acc = make_float4(0.f, 0.f, 0.f, 0.f);

        if (deg > 0) {
            // Stage chunk 0 (clamp addresses inside this node's segment)
            {
                int idx = beg + lane; if (idx >= end) idx = end - 1;
                const int* g = bucket + idx;
                asm volatile("global_load_async_to_lds_b32 %0, %1, off"
                             :: "v"(lds0), "v"(g) : "memory");
            }
            const int nchunks = (deg + 31) >> 5;
            for (int ch = 0; ch < nchunks; ++ch) {
                if (ch + 1 < nchunks) {
                    // Prefetch next chunk into the other buffer, then wait for
                    // the in-flight older one (async loads complete in order).
                    int idx = beg + (ch + 1) * 32 + lane; if (idx >= end) idx = end - 1;
                    const int* g = bucket + idx;
                    unsigned dstl = ((ch + 1) & 1) ? lds1 : lds0;
                    asm volatile("global_load_async_to_lds_b32 %0, %1, off"
                                 :: "v"(dstl), "v"(g) : "memory");
                    asm volatile("s_wait_asynccnt 0x1" ::: "memory");
                } else {
                    asm volatile("s_wait_asynccnt 0x0" ::: "memory");
                }
                int cnt = deg - ch * 32; if (cnt > 32) cnt = 32;
                const int* sc = s_cols[wave][ch & 1];
                #pragma unroll 4
                for (int k = 0; k < cnt; ++k) {
                    int c = sc[k];             // wave-uniform -> ds broadcast
                    const float4 v = *reinterpret_cast<const float4*>(
                        x + (size_t)c * N_FEAT + lane * 4);   // global_load_b128, L2 hit
                    acc.x += v.x; acc.y += v.y; acc.z += v.z; acc.w += v.w;
                }
            }
        }
        // Exactly one store per output element; no prior zeroing needed.
        *reinterpret_cast<float4*>(out + (size_t)node * N_FEAT + lane * 4) = acc;
    }
}

// ---------------------------------------------------------------------------
// Fallback: atomic scatter (used only if d_ws cannot hold the CSR scratch)
// ---------------------------------------------------------------------------

__global__ __launch_bounds__(256) void zero_out_kernel(float* __restrict__ out, int n4) {
    float4 z = make_float4(0.f, 0.f, 0.f, 0.f);
    for (int i = blockIdx.x * blockDim.x + threadIdx.x; i < n4;
         i += gridDim.x * blockDim.x)
        reinterpret_cast<float4*>(out)[i] = z;
}

__global__ __launch_bounds__(256) void scatter_add_kernel(
        const float* __restrict__ x,
        const int*   __restrict__ rows,
        const int*   __restrict__ cols,
        float*       __restrict__ out,
        int n_edges) {
    __shared__ int s_row[TILE];
    __shared__ int s_col[TILE];

    const int tid  = threadIdx.x;
    const int lane = tid & 31;
    const int wave = tid >> 5;
    const int base = blockIdx.x * TILE;

    {   // async-stage this tile's indices (each wave stages what it consumes)
        int e  = base + tid;
        int ec = (e < n_edges) ? e : (n_edges - 1);
        unsigned lds_r = (unsigned)(size_t)&s_row[tid];
        unsigned lds_c = (unsigned)(size_t)&s_col[tid];
        const int* gr = rows + ec;
        const int* gc = cols + ec;
        asm volatile("global_load_async_to_lds_b32 %0, %1, off"
                     :: "v"(lds_r), "v"(gr) : "memory");
        asm volatile("global_load_async_to_lds_b32 %0, %1, off"
                     :: "v"(lds_c), "v"(gc) : "memory");
        asm volatile("s_wait_asynccnt 0x0" ::: "memory");
    }
    __syncthreads();

    const int estart = wave * (TILE / 8);
    const int eend   = estart + (TILE / 8);
    #pragma unroll 2
    for (int i = estart; i < eend; ++i) {
        int e = base + i;
        if (e >= n_edges) break;
        int r = s_row[i];
        int c = s_col[i];
        const float4 v = *reinterpret_cast<const float4*>(
            x + (size_t)c * N_FEAT + lane * 4);
        float* dst = out + (size_t)r * N_FEAT + lane * 4;
        asm volatile(
            "global_atomic_add_f32 %0, %1, off\n\t"
            "global_atomic_add_f32 %0, %2, off offset:4\n\t"
            "global_atomic_add_f32 %0, %3, off offset:8\n\t"
            "global_atomic_add_f32 %0, %4, off offset:12"
            :: "v"(dst), "v"(v.x), "v"(v.y), "v"(v.z), "v"(v.w)
            : "memory");
    }
}

// ---------------------------------------------------------------------------
// Host launcher
// ---------------------------------------------------------------------------

extern "C" void kernel_launch(void* const* d_in, const int* in_sizes, int n_in,
                              void* d_out, int out_size, void* d_ws, size_t ws_size,
                              hipStream_t stream) {
    const float* x    = (const float*)d_in[0];
    const int*   edge = (const int*)d_in[1];      // [2, E] flat
    const int    E    = in_sizes[1] / 2;
    const int    N    = out_size / N_FEAT;        // 10000

    const int* rows = edge;
    const int* cols = edge + E;
    float*     out  = (float*)d_out;

    // Workspace layout: cnt/cursor [N] | offs [N+1] | bucket [E]
    const size_t needed = ((size_t)N + (size_t)(N + 1) + (size_t)E) * sizeof(int);

    if (ws_size >= needed) {
        int* cnt    = (int*)d_ws;          // later rewritten as cursor by scan
        int* offs   = cnt + N;
        int* bucket = offs + N + 1;

        int zb = (N + 255) / 256;
        int eb = (E + 255) / 256;
        int gb = (N + WPB - 1) / WPB;

        zero_cnt_kernel  <<<zb,   256, 0, stream>>>(cnt, N);
        count_kernel     <<<eb,   256, 0, stream>>>(rows, cnt, E);
        scan_kernel      <<<1,   1024, 0, stream>>>(cnt, offs, N, E);
        bucket_kernel    <<<eb,   256, 0, stream>>>(rows, cols, cnt, bucket, E);
        gather_sum_kernel<<<gb,   256, 0, stream>>>(x, offs, bucket, out, N);
    } else {
        // Fallback: zero + atomic scatter-add
        int n4 = out_size / 4;
        zero_out_kernel   <<<(n4 + 255) / 256, 256, 0, stream>>>(out, n4);
        scatter_add_kernel<<<(E + TILE - 1) / TILE, 256, 0, stream>>>(x, rows, cols, out, E);
    }
}